// WSM_74500502716670
// MI455X (gfx1250) — compile-verified
//
#include <hip/hip_runtime.h>
#include <hip/hip_bf16.h>
#include <math.h>

typedef __attribute__((ext_vector_type(2))) float v2f;
typedef __attribute__((ext_vector_type(8))) float v8f;

#define SM_L   6368
#define SM_H1  2133
#define SM_H2  4250
#define SM_T   128
#define SM_B   4
#define SM_D   16
#define SM_KS  (SM_L / 8)      /* 796: per-wave K slice in step kernel */

// ---------------------------------------------------------------------------
// Recurrence step: theta_out[b][m] = sum_k A[m][k]*theta_in[b][k]
//                                  + sum_j Bm[m][j]*dif[b][j]
// One block per 16-row tile (6368 = 398*16 exactly). 8 waves split K 8 ways,
// accumulate on V_WMMA_F32_16X16X4_F32, combine partials through LDS.
//
// B-fragment trick: WMMA columns are independent (D[:,n] depends only on
// B[:,n]), and we only read accumulator columns 0-3, so lanes 4-15 load the
// clamped column (l16 & 3) unconditionally instead of predicating + zeroing.
// Garbage lands in accumulator columns we never store. No exec masking in the
// hot loop.
// ---------------------------------------------------------------------------
__global__ __launch_bounds__(256)
void WSM_step(const float* __restrict__ Am, const float* __restrict__ x,
              const float* __restrict__ Bm, const float* __restrict__ dif,
              float* __restrict__ out)
{
    __shared__ float part[8][64];

    const int lane = threadIdx.x & 31;
    const int wave = threadIdx.x >> 5;
    const int row0 = blockIdx.x << 4;
    const int half = lane >> 4;          // 0: K0/K1 half, 1: K2/K3 half
    const int l16  = lane & 15;
    const int k0   = wave * SM_KS;

    const float* __restrict__ wrow = Am + (size_t)(row0 + l16) * SM_L + k0 + 2 * half;
    const float* __restrict__ xcol = x  + (size_t)(l16 & 3) * SM_L + k0 + 2 * half;

    v8f acc0 = {0.f,0.f,0.f,0.f,0.f,0.f,0.f,0.f};
    v8f acc1 = {0.f,0.f,0.f,0.f,0.f,0.f,0.f,0.f};

    const int kfull = SM_KS & ~7;        // 792
    #pragma unroll 4
    for (int k = 0; k < kfull; k += 8) {
        if ((k & 255) == 0)
            __builtin_prefetch(wrow + k + 1024, 0, 1);   // stream-ahead on A
        v2f a0 = *(const v2f*)(wrow + k);
        v2f a1 = *(const v2f*)(wrow + k + 4);
        v2f b0 = *(const v2f*)(xcol + k);
        v2f b1 = *(const v2f*)(xcol + k + 4);
        acc0 = __builtin_amdgcn_wmma_f32_16x16x4_f32(false, a0, false, b0,
                                                     (short)0, acc0, false, false);
        acc1 = __builtin_amdgcn_wmma_f32_16x16x4_f32(false, a1, false, b1,
                                                     (short)0, acc1, false, false);
    }
    for (int k = kfull; k < SM_KS; k += 4) {             // one K=4 leftover
        v2f a0 = *(const v2f*)(wrow + k);
        v2f b0 = *(const v2f*)(xcol + k);
        acc0 = __builtin_amdgcn_wmma_f32_16x16x4_f32(false, a0, false, b0,
                                                     (short)0, acc0, false, false);
    }

    // C layout: vgpr r -> row (half*8 + r) of the tile, batch = l16 (<4).
    if (l16 < 4) {
        #pragma unroll
        for (int r = 0; r < 8; ++r)
            part[wave][(half * 8 + r) * 4 + l16] = acc0[r] + acc1[r];
    }
    __syncthreads();

    const int t = threadIdx.x;
    if (t < 64) {
        const int b  = t & 3;
        const int m  = row0 + (t >> 2);
        float v = 0.f;
        #pragma unroll
        for (int w = 0; w < 8; ++w) v += part[w][t];
        const float* __restrict__ bm = Bm + (size_t)m * 17;
        const float* __restrict__ db = dif + b * 17;
        #pragma unroll
        for (int j = 0; j < 17; ++j) v += bm[j] * db[j];
        out[(size_t)b * SM_L + m] = v;
    }
}

// ---------------------------------------------------------------------------
// Generic batched (N=4) GEMV on WMMA f32 16x16x4 tiles (encoder stages only:
// ragged M/K, optional bias + relu). One wave owns one 16-row tile, full K.
// Same clamped-column branchless B loads as WSM_step.
// ---------------------------------------------------------------------------
__global__ __launch_bounds__(256)
void WSM_gemv16(const float* __restrict__ W, const float* __restrict__ x,
                const float* __restrict__ bias, float* __restrict__ out,
                int M, int K, int relu)
{
    const int lane = threadIdx.x & 31;
    const int wave = threadIdx.x >> 5;
    const int tile = blockIdx.x * 8 + wave;
    const int ntiles = (M + 15) >> 4;
    if (tile >= ntiles) return;           // wave-uniform: EXEC all-ones inside

    const int row0 = tile << 4;
    const int half = lane >> 4;
    const int l16  = lane & 15;
    int rA = row0 + l16; if (rA > M - 1) rA = M - 1;   // clamped rows never stored
    const float* __restrict__ wrow = W + (size_t)rA * K + 2 * half;
    const float* __restrict__ xcol = x + (size_t)(l16 & 3) * K + 2 * half;

    v8f acc0 = {0.f,0.f,0.f,0.f,0.f,0.f,0.f,0.f};
    v8f acc1 = {0.f,0.f,0.f,0.f,0.f,0.f,0.f,0.f};
    const int kmax  = K & ~3;
    const int kmax8 = K & ~7;

    #pragma unroll 4
    for (int k = 0; k < kmax8; k += 8) {
        if ((k & 255) == 0)
            __builtin_prefetch(wrow + k + 2048, 0, 1);
        v2f a0 = *(const v2f*)(wrow + k);
        v2f a1 = *(const v2f*)(wrow + k + 4);
        v2f b0 = *(const v2f*)(xcol + k);
        v2f b1 = *(const v2f*)(xcol + k + 4);
        acc0 = __builtin_amdgcn_wmma_f32_16x16x4_f32(false, a0, false, b0,
                                                     (short)0, acc0, false, false);
        acc1 = __builtin_amdgcn_wmma_f32_16x16x4_f32(false, a1, false, b1,
                                                     (short)0, acc1, false, false);
    }
    if (kmax8 < kmax) {
        v2f a0 = *(const v2f*)(wrow + kmax8);
        v2f b0 = *(const v2f*)(xcol + kmax8);
        acc0 = __builtin_amdgcn_wmma_f32_16x16x4_f32(false, a0, false, b0,
                                                     (short)0, acc0, false, false);
    }
    v8f acc;
    #pragma unroll
    for (int r = 0; r < 8; ++r) acc[r] = acc0[r] + acc1[r];

    if (l16 < 4) {
        const int b = l16;
        const int mbase = row0 + half * 8;
        const float* __restrict__ xb = x + (size_t)b * K;
        #pragma unroll
        for (int r = 0; r < 8; ++r) {
            int m = mbase + r;
            if (m >= M) break;
            float v = acc[r];
            const float* __restrict__ wr = W + (size_t)m * K;
            for (int k = kmax; k < K; ++k)       // scalar K tail (K%4)
                v += wr[k] * xb[k];
            if (bias) v += bias[m];
            if (relu) v = fmaxf(v, 0.f);
            out[(size_t)b * M + m] = v;
        }
    }
}

// ---------------------------------------------------------------------------
// Init: pack x0[b][16], xhat = xs[:,0,:], diff_0 = 0 (exact: t_prev=t_0 and
// x_prev=x_t=xs[:,0] at step 0).
// ---------------------------------------------------------------------------
__global__ void WSM_init(const float* __restrict__ xs, float* __restrict__ x0,
                         float* __restrict__ xhat, float* __restrict__ dif)
{
    int t = threadIdx.x;
    if (t < SM_B * SM_D) {
        int b = t >> 4, j = t & 15;
        float v = xs[(size_t)b * SM_T * SM_D + j];
        x0[b * SM_D + j]   = v;
        xhat[b * SM_D + j] = v;
    }
    if (t < SM_B * 17) dif[t] = 0.f;
}

// ---------------------------------------------------------------------------
// Per-step tiny MLP (root_apply) + recurrence bookkeeping. theta = theta_{i+1}.
// One wave per batch. tin = 2*t_i - t_{i-1}.
// ---------------------------------------------------------------------------
__global__ __launch_bounds__(128)
void WSM_mlp(const float* __restrict__ theta, const float* __restrict__ xs,
             const float* __restrict__ ts, float* __restrict__ xhat,
             float* __restrict__ dif, float* __restrict__ out, int i)
{
    __shared__ float h0s[SM_B][64];
    __shared__ float h1s[SM_B][64];
    const int b    = threadIdx.x >> 5;
    const int lane = threadIdx.x & 31;
    const float* __restrict__ th = theta + (size_t)b * SM_L;

    const float t_i    = ts[b * SM_T + i];
    const float t_prev = (i > 0) ? ts[b * SM_T + i - 1] : t_i;
    const float tin    = 2.f * t_i - t_prev;

    for (int r = lane; r < 64; r += 32)
        h0s[b][r] = fmaxf(th[r] * tin + th[64 + r], 0.f);
    __syncthreads();

    for (int r = lane; r < 64; r += 32) {
        const float* __restrict__ w = th + 128 + r * 64;
        float a = th[4224 + r];
        #pragma unroll 8
        for (int c = 0; c < 64; ++c) a += w[c] * h0s[b][c];
        h1s[b][r] = fmaxf(a, 0.f);
    }
    __syncthreads();

    {
        const float* __restrict__ w = th + 4288 + lane * 64;
        float a = th[6336 + lane];
        #pragma unroll 8
        for (int c = 0; c < 64; ++c) a += w[c] * h1s[b][c];
        float res;
        if (lane < SM_D) res = tanhf(a);
        else             res = fmaxf(a, 0.f) + log1pf(expf(-fabsf(a)));  // stable softplus
        out[((size_t)b * SM_T + i) * 32 + lane] = res;

        if (lane < SM_D) {
            float xh_old = xhat[b * SM_D + lane];      // = xhat_i
            if (i + 1 < SM_T) {
                dif[b * 17 + 1 + lane] =
                    xs[((size_t)b * SM_T + (i + 1)) * SM_D + lane] - xh_old;
                if (lane == 0)
                    dif[b * 17] = ts[b * SM_T + i + 1] - t_i;
            }
            xhat[b * SM_D + lane] = res;               // xhat_{i+1} = tanh(mean)
        }
    }
}

// ---------------------------------------------------------------------------
extern "C" void kernel_launch(void* const* d_in, const int* in_sizes, int n_in,
                              void* d_out, int out_size, void* d_ws, size_t ws_size,
                              hipStream_t stream)
{
    (void)in_sizes; (void)n_in; (void)out_size; (void)ws_size;
    const float* xs = (const float*)d_in[0];
    const float* ts = (const float*)d_in[1];
    const float* A  = (const float*)d_in[2];
    const float* Bm = (const float*)d_in[3];
    const float* W1 = (const float*)d_in[4];
    const float* b1 = (const float*)d_in[5];
    const float* W2 = (const float*)d_in[6];
    const float* b2 = (const float*)d_in[7];
    const float* W3 = (const float*)d_in[8];
    const float* b3 = (const float*)d_in[9];
    float* out = (float*)d_out;

    float* ws     = (float*)d_ws;
    float* thbuf0 = ws;                       // [4][L]
    float* thbuf1 = thbuf0 + SM_B * SM_L;     // [4][L]
    float* h1v    = thbuf1 + SM_B * SM_L;     // [4][H1]
    float* h2v    = h1v    + SM_B * SM_H1;    // [4][H2]
    float* x0v    = h2v    + SM_B * SM_H2;    // [4][16]
    float* xhat   = x0v + SM_B * SM_D;        // [4][16]
    float* dif    = xhat + SM_B * SM_D;       // [4][17]

    auto grid_for = [](int M) { return ((((M + 15) >> 4) + 7) >> 3); };

    WSM_init<<<1, 128, 0, stream>>>(xs, x0v, xhat, dif);

    // Encoder: theta0 = W3 @ relu(W2 @ relu(W1 @ x0 + b1) + b2) + b3
    WSM_gemv16<<<grid_for(SM_H1), 256, 0, stream>>>(W1, x0v, b1, h1v, SM_H1, SM_D, 1);
    WSM_gemv16<<<grid_for(SM_H2), 256, 0, stream>>>(W2, h1v, b2, h2v, SM_H2, SM_H1, 1);
    WSM_gemv16<<<grid_for(SM_L),  256, 0, stream>>>(W3, h2v, b3, thbuf0, SM_L, SM_H2, 0);

    // Recurrence: theta_{i+1} = A @ theta_i + Bm @ diff_i ; then root MLP.
    for (int i = 0; i < SM_T; ++i) {
        const float* tc = (i & 1) ? thbuf1 : thbuf0;
        float*       tn = (i & 1) ? thbuf0 : thbuf1;
        WSM_step<<<SM_L / 16, 256, 0, stream>>>(A, tc, Bm, dif, tn);
        WSM_mlp<<<1, 128, 0, stream>>>(tn, xs, ts, xhat, dif, out, i);
    }
}